// Network_39625368273588
// MI455X (gfx1250) — compile-verified
//
#include <hip/hip_runtime.h>
#include <math.h>

typedef __attribute__((ext_vector_type(2))) float v2f;
typedef __attribute__((ext_vector_type(8))) float v8f;

#define NBATCH 4
#define NW_    1536
#define NF_    16
#define NCROSS 400000
#define NEDGES 1600000
#define INDIM  34      // 2*NF + 2
#define LRELU  0.2f

// ---------------------------------------------------------------- utilities

__device__ __forceinline__ void atomicMaxF(float* addr, float v) {
    // sign-split trick: word initialized to -inf (0xFF800000)
    if (v >= 0.0f) atomicMax((int*)addr, __float_as_int(v));
    else           atomicMin((unsigned int*)addr, (unsigned int)__float_as_int(v));
}

__device__ __forceinline__ float getx(const float* __restrict__ fw,
                                      const float* __restrict__ sw,
                                      float xc0, float xc1, int k) {
    if (k < 16) return fw[k];
    if (k < 32) return sw[k - 16];
    if (k == 32) return xc0;
    if (k == 33) return xc1;
    return 0.0f;
}

__device__ __forceinline__ float getw(const float* __restrict__ W,
                                      float a0, float a1, float a2, float a3, int k) {
    if (k >= INDIM) return 0.0f;
    const float* w = W + k * 4;
    return w[0]*a0 + w[1]*a1 + w[2]*a2 + w[3]*a3;
}

// ---------------------------------------------------------------- init

__global__ void k_init(float* __restrict__ M, float* __restrict__ DEN,
                       float* __restrict__ YACC, int n) {
    int i = blockIdx.x * blockDim.x + threadIdx.x;
    if (i < n) { M[i] = -INFINITY; DEN[i] = 0.0f; YACC[i] = 0.0f; }
}

// a_e[e] = edge_attr[e] . (W_e @ att_edge)
__global__ void k_ae(const float* __restrict__ edge_attr,
                     const float* __restrict__ We,
                     const float* __restrict__ att_edge,
                     float* __restrict__ AE) {
    int e = blockIdx.x * blockDim.x + threadIdx.x;
    if (e >= NEDGES) return;
    float c0 = We[0]*att_edge[0]  + We[1]*att_edge[1]  + We[2]*att_edge[2]  + We[3]*att_edge[3];
    float c1 = We[4]*att_edge[0]  + We[5]*att_edge[1]  + We[6]*att_edge[2]  + We[7]*att_edge[3];
    float c2 = We[8]*att_edge[0]  + We[9]*att_edge[1]  + We[10]*att_edge[2] + We[11]*att_edge[3];
    float c3 = We[12]*att_edge[0] + We[13]*att_edge[1] + We[14]*att_edge[2] + We[15]*att_edge[3];
    const float4 ea = ((const float4*)edge_attr)[e];
    AE[e] = ea.x*c0 + ea.y*c1 + ea.z*c2 + ea.w*c3;
}

// ---------------------------------------------------------------- phase 1: WMMA GEMM
// Per wave: one 16-row tile of crossings, K=36 (9 x WMMA f32 16x16x4), N=16 (cols 0..2 live).
// Ssrc[i*4+b] = x_i . (W@att_src), Sdst = . (W@att_dst), P = . (W@w_mlp)

__global__ __launch_bounds__(256) void k_feat(
    const float* __restrict__ fwires, const float* __restrict__ swires,
    const float* __restrict__ crossings, const float* __restrict__ bbox,
    const int*   __restrict__ indices,  const float* __restrict__ W,
    const float* __restrict__ att_src,  const float* __restrict__ att_dst,
    const float* __restrict__ w_mlp,
    float* __restrict__ Ssrc, float* __restrict__ Sdst, float* __restrict__ P)
{
    const int lane  = threadIdx.x & 31;
    const int wave  = threadIdx.x >> 5;
    const int tile  = blockIdx.x * 8 + wave;      // 25000 tiles exactly (400000/16)
    const int b     = blockIdx.y;
    const int mrow  = lane & 15;                  // A row within tile / B,C column
    const int khalf = (lane < 16) ? 0 : 2;        // K sub-offset per ISA f32 A/B layout
    const int row   = tile * 16 + mrow;           // crossing index

    const float bb0 = bbox[0], bb1 = bbox[1], bb2 = bbox[2];
    const float invb = rsqrtf(bb0*bb0 + bb1*bb1 + bb2*bb2);

    const int i0 = indices[row * 2 + 0];
    const int i1 = indices[row * 2 + 1];
    const float* fw = fwires + (size_t)(b * NW_ + i0) * NF_;   // T == 1
    const float* sw = swires + (size_t)(b * NW_ + i1) * NF_;
    const float xc0 = crossings[row * 2 + 0] * invb;
    const float xc1 = crossings[row * 2 + 1] * invb;

    // column vector of the augmented weight matrix for this lane's N column
    float a0 = 0.f, a1 = 0.f, a2 = 0.f, a3 = 0.f;
    if      (mrow == 0) { a0 = att_src[0]; a1 = att_src[1]; a2 = att_src[2]; a3 = att_src[3]; }
    else if (mrow == 1) { a0 = att_dst[0]; a1 = att_dst[1]; a2 = att_dst[2]; a3 = att_dst[3]; }
    else if (mrow == 2) { a0 = w_mlp[0];   a1 = w_mlp[1];   a2 = w_mlp[2];   a3 = w_mlp[3];   }

    v8f acc = {0.f, 0.f, 0.f, 0.f, 0.f, 0.f, 0.f, 0.f};
    #pragma unroll
    for (int c = 0; c < 9; ++c) {
        const int k0 = 4 * c + khalf;
        v2f A  = { getx(fw, sw, xc0, xc1, k0), getx(fw, sw, xc0, xc1, k0 + 1) };
        v2f Bv = { getw(W, a0, a1, a2, a3, k0), getw(W, a0, a1, a2, a3, k0 + 1) };
        acc = __builtin_amdgcn_wmma_f32_16x16x4_f32(
                  false, A, false, Bv, (short)0, acc, false, false);
    }

    // C/D layout: VGPR j -> M=j (lanes 0-15) / M=j+8 (lanes 16-31), N = lane%16
    if (mrow < 3) {
        float* T = (mrow == 0) ? Ssrc : (mrow == 1) ? Sdst : P;
        const int rbase = tile * 16 + ((lane < 16) ? 0 : 8);
        #pragma unroll
        for (int j = 0; j < 8; ++j)
            T[(size_t)(rbase + j) * 4 + b] = acc[j];
    }
}

// ---------------------------------------------------------------- edge passes (all 4 batches fused)

__device__ __forceinline__ float4 edge_logits(const float* __restrict__ Ssrc,
                                              const float* __restrict__ Sdst,
                                              float ae, int s, int d) {
    const float4 hs = ((const float4*)Ssrc)[s];
    const float4 hd = ((const float4*)Sdst)[d];
    float4 z;
    z.x = hs.x + hd.x + ae; z.y = hs.y + hd.y + ae;
    z.z = hs.z + hd.z + ae; z.w = hs.w + hd.w + ae;
    z.x = (z.x > 0.f) ? z.x : LRELU * z.x;
    z.y = (z.y > 0.f) ? z.y : LRELU * z.y;
    z.z = (z.z > 0.f) ? z.z : LRELU * z.z;
    z.w = (z.w > 0.f) ? z.w : LRELU * z.w;
    return z;
}

__global__ void k_edge_max(const int* __restrict__ ei, const float* __restrict__ AE,
                           const float* __restrict__ Ssrc, const float* __restrict__ Sdst,
                           float* __restrict__ M) {
    int e = blockIdx.x * blockDim.x + threadIdx.x;
    if (e >= NEDGES) return;
    const int s = ei[e], d = ei[NEDGES + e];
    const float4 z = edge_logits(Ssrc, Sdst, AE[e], s, d);
    atomicMaxF(&M[(size_t)d * 4 + 0], z.x);
    atomicMaxF(&M[(size_t)d * 4 + 1], z.y);
    atomicMaxF(&M[(size_t)d * 4 + 2], z.z);
    atomicMaxF(&M[(size_t)d * 4 + 3], z.w);
}

__global__ void k_edge_den(const int* __restrict__ ei, const float* __restrict__ AE,
                           const float* __restrict__ Ssrc, const float* __restrict__ Sdst,
                           const float* __restrict__ M, float* __restrict__ DEN) {
    int e = blockIdx.x * blockDim.x + threadIdx.x;
    if (e >= NEDGES) return;
    const int s = ei[e], d = ei[NEDGES + e];
    const float4 z = edge_logits(Ssrc, Sdst, AE[e], s, d);
    const float4 m = ((const float4*)M)[d];
    atomicAdd(&DEN[(size_t)d * 4 + 0], expf(z.x - m.x));
    atomicAdd(&DEN[(size_t)d * 4 + 1], expf(z.y - m.y));
    atomicAdd(&DEN[(size_t)d * 4 + 2], expf(z.z - m.z));
    atomicAdd(&DEN[(size_t)d * 4 + 3], expf(z.w - m.w));
}

__global__ void k_edge_acc(const int* __restrict__ ei, const float* __restrict__ AE,
                           const float* __restrict__ Ssrc, const float* __restrict__ Sdst,
                           const float* __restrict__ M, const float* __restrict__ DEN,
                           const float* __restrict__ P, float* __restrict__ YACC) {
    int e = blockIdx.x * blockDim.x + threadIdx.x;
    if (e >= NEDGES) return;
    const int s = ei[e], d = ei[NEDGES + e];
    const float4 z  = edge_logits(Ssrc, Sdst, AE[e], s, d);
    const float4 m  = ((const float4*)M)[d];
    const float4 dn = ((const float4*)DEN)[d];
    const float4 p  = ((const float4*)P)[s];
    atomicAdd(&YACC[(size_t)d * 4 + 0], expf(z.x - m.x) / (dn.x + 1e-16f) * p.x);
    atomicAdd(&YACC[(size_t)d * 4 + 1], expf(z.y - m.y) / (dn.y + 1e-16f) * p.y);
    atomicAdd(&YACC[(size_t)d * 4 + 2], expf(z.z - m.z) / (dn.z + 1e-16f) * p.z);
    atomicAdd(&YACC[(size_t)d * 4 + 3], expf(z.w - m.w) / (dn.w + 1e-16f) * p.w);
}

// ---------------------------------------------------------------- final: y[b][i] = YACC[i][b] + (bias.w_mlp + b_mlp)

__global__ void k_final(const float* __restrict__ YACC,
                        const float* __restrict__ bias, const float* __restrict__ w_mlp,
                        const float* __restrict__ b_mlp, float* __restrict__ out) {
    int i = blockIdx.x * blockDim.x + threadIdx.x;
    if (i >= NCROSS) return;
    const float cconst = bias[0]*w_mlp[0] + bias[1]*w_mlp[1] +
                         bias[2]*w_mlp[2] + bias[3]*w_mlp[3] + b_mlp[0];
    const float4 y = ((const float4*)YACC)[i];
    out[(size_t)0 * NCROSS + i] = y.x + cconst;
    out[(size_t)1 * NCROSS + i] = y.y + cconst;
    out[(size_t)2 * NCROSS + i] = y.z + cconst;
    out[(size_t)3 * NCROSS + i] = y.w + cconst;
}

// ---------------------------------------------------------------- launch

extern "C" void kernel_launch(void* const* d_in, const int* in_sizes, int n_in,
                              void* d_out, int out_size, void* d_ws, size_t ws_size,
                              hipStream_t stream) {
    const float* fwires    = (const float*)d_in[0];
    const float* swires    = (const float*)d_in[1];
    const float* crossings = (const float*)d_in[2];
    const float* bbox      = (const float*)d_in[3];
    const float* edge_attr = (const float*)d_in[4];
    const int*   indices   = (const int*)  d_in[5];
    const int*   edge_idx  = (const int*)  d_in[6];
    const float* W         = (const float*)d_in[7];
    const float* att_src   = (const float*)d_in[8];
    const float* att_dst   = (const float*)d_in[9];
    const float* W_e       = (const float*)d_in[10];
    const float* att_edge  = (const float*)d_in[11];
    const float* bias      = (const float*)d_in[12];
    const float* w_mlp     = (const float*)d_in[13];
    const float* b_mlp     = (const float*)d_in[14];
    float* out = (float*)d_out;

    float* ws   = (float*)d_ws;
    float* Ssrc = ws;                                 // NCROSS*4
    float* Sdst = Ssrc + (size_t)NCROSS * 4;
    float* P    = Sdst + (size_t)NCROSS * 4;
    float* M    = P    + (size_t)NCROSS * 4;
    float* DEN  = M    + (size_t)NCROSS * 4;
    float* YACC = DEN  + (size_t)NCROSS * 4;
    float* AE   = YACC + (size_t)NCROSS * 4;          // NEDGES

    const int TB = 256;
    k_init<<<(NCROSS * 4 + TB - 1) / TB, TB, 0, stream>>>(M, DEN, YACC, NCROSS * 4);
    k_ae  <<<(NEDGES + TB - 1) / TB, TB, 0, stream>>>(edge_attr, W_e, att_edge, AE);

    dim3 gfeat(NCROSS / 16 / 8, NBATCH);              // 3125 x 4 blocks, 8 waves each
    k_feat<<<gfeat, TB, 0, stream>>>(fwires, swires, crossings, bbox, indices, W,
                                     att_src, att_dst, w_mlp, Ssrc, Sdst, P);

    const int ge = (NEDGES + TB - 1) / TB;
    k_edge_max<<<ge, TB, 0, stream>>>(edge_idx, AE, Ssrc, Sdst, M);
    k_edge_den<<<ge, TB, 0, stream>>>(edge_idx, AE, Ssrc, Sdst, M, DEN);
    k_edge_acc<<<ge, TB, 0, stream>>>(edge_idx, AE, Ssrc, Sdst, M, DEN, P, YACC);

    k_final<<<(NCROSS + TB - 1) / TB, TB, 0, stream>>>(YACC, bias, w_mlp, b_mlp, out);
}